// DielectricReadout_28329604285242
// MI455X (gfx1250) — compile-verified
//
#include <hip/hip_runtime.h>
#include <math.h>

#define N_NODES 200000
#define DIM     256
#define HID     1024
#define OUTC    4002
#define LHALF   2001
#define NGRAPH  4096

#define KC      64             // K-chunk staged in LDS (double-buffered)
#define MT      4              // row tiles (16 rows each) per wave

typedef __attribute__((ext_vector_type(2))) float v2f;
typedef __attribute__((ext_vector_type(8))) float v8f;

#if __has_builtin(__builtin_amdgcn_global_load_async_to_lds_b128)
#define HAVE_ASYNC_LDS 1
typedef __attribute__((__vector_size__(4 * sizeof(int)))) int vi4;
typedef __attribute__((address_space(1))) vi4 gvi4;   // global (device) AS
typedef __attribute__((address_space(3))) vi4 lvi4;   // LDS AS
#else
#define HAVE_ASYNC_LDS 0
#endif

__device__ __forceinline__ void wait_async_lds() {
#if HAVE_ASYNC_LDS
#if __has_builtin(__builtin_amdgcn_s_wait_asynccnt)
    __builtin_amdgcn_s_wait_asynccnt(0);
#else
    asm volatile("s_wait_asynccnt 0x0" ::: "memory");
#endif
#endif
}

// ---------------------------------------------------------------------------
// Segment boundaries from sorted node_graph_index: seg[g] = lower_bound(idx, g)
// ---------------------------------------------------------------------------
__global__ void seg_bounds_kernel(const int* __restrict__ idx, int* __restrict__ seg) {
    int g = blockIdx.x * blockDim.x + threadIdx.x;
    if (g > NGRAPH) return;
    if (g == NGRAPH) { seg[g] = N_NODES; return; }
    int lo = 0, hi = N_NODES;
    while (lo < hi) {
        int mid = (lo + hi) >> 1;
        if (idx[mid] < g) lo = mid + 1; else hi = mid;
    }
    seg[g] = lo;
}

// ---------------------------------------------------------------------------
// fp32 WMMA GEMM, compile-time shapes: C = act(A[M,K] @ W[K,NC] + bias)
//  - block: 256 threads = 8 waves; block tile = 64 rows x 128 cols
//  - each wave: MT=4 row tiles x 1 col tile (B fragment reused 4x)
//  - A panel 64 x KC double-buffered in LDS, filled with async-to-LDS b128,
//    next chunk's fill overlapped with current chunk's WMMAs (ASYNCcnt)
//  - MASK=false: NC multiple of 128, no guards at all
//  - MODE 0: SiLU -> C ; MODE 1: bias only, fused out/eps_imag/eps_real stores
// ---------------------------------------------------------------------------
template<int K, int NC, int MODE, bool MASK>
__global__ __launch_bounds__(256) void gemm_wmma_kernel(
    const float* __restrict__ A, const float* __restrict__ W,
    const float* __restrict__ bias, float* __restrict__ C,
    float* __restrict__ eimag, float* __restrict__ ereal)
{
    constexpr int LDA    = KC + 4;       // word stride 68: conflict-free, 16B-aligned
    constexpr int NCHUNK = K / KC;
    __shared__ float Alds[2][64 * LDA];

    const int tid    = threadIdx.x;
    const int lane   = tid & 31;        // wave32
    const int laneLo = lane & 15;
    const int hi     = lane >> 4;       // upper half-wave: K+2/K+3, rows +8
    const int wave   = tid >> 5;
    const int r0     = blockIdx.x * 64;
    const int c0     = (blockIdx.y * 8 + wave) * 16;
    const int c      = c0 + laneLo;
    const bool cok   = !MASK || (c < NC);
    const int  cc    = MASK ? (c < NC ? c : NC - 1) : c;   // clamped (always valid)

    v8f acc[MT];
    #pragma unroll
    for (int t = 0; t < MT; ++t) acc[t] = (v8f){0.f,0.f,0.f,0.f,0.f,0.f,0.f,0.f};

    // per-lane B base: row (hi*2), clamped column
    const float* __restrict__ wbase = W + (size_t)(hi * 2) * NC + cc;

    // ---- panel fill: 64 x KC floats, 4 x b128 per thread -------------------
    auto fill_panel = [&](int kc, int b) {
        #pragma unroll
        for (int i = tid; i < 64 * (KC / 4); i += 256) {
            int row  = i >> 4;              // / (KC/4) with KC==64
            int col4 = (i & 15) * 4;
#if HAVE_ASYNC_LDS
            __builtin_amdgcn_global_load_async_to_lds_b128(
                (gvi4*)(A + (size_t)(r0 + row) * K + kc + col4),
                (lvi4*)&Alds[b][row * LDA + col4],
                0, 0);
#else
            float4 v = *(const float4*)&A[(size_t)(r0 + row) * K + kc + col4];
            *(float4*)&Alds[b][row * LDA + col4] = v;
#endif
        }
    };

    // ---- software pipeline: fill(ch+1) overlapped with compute(ch) ---------
    fill_panel(0, 0);
    wait_async_lds();
    __syncthreads();

    int buf = 0;
    for (int ch = 0; ch < NCHUNK; ++ch) {
        if (ch + 1 < NCHUNK) fill_panel((ch + 1) * KC, buf ^ 1);

        const float* __restrict__ wp = wbase + (size_t)(ch * KC) * NC;
        const float* __restrict__ ap = &Alds[buf][laneLo * LDA + hi * 2];
        #pragma unroll 8
        for (int ks = 0; ks < KC; ks += 4) {
            // B fragment: rows ch*KC+ks+hi*2 and +1, column cc (const offsets)
            float bx = wp[(size_t)ks * NC];
            float by = wp[(size_t)(ks + 1) * NC];
            if (MASK) { bx = cok ? bx : 0.f; by = cok ? by : 0.f; }
            v2f b; b.x = bx; b.y = by;
            // 4 row tiles share this B fragment
            #pragma unroll
            for (int t = 0; t < MT; ++t) {
                const v2f a = *(const v2f*)&ap[t * 16 * LDA + ks];
                acc[t] = __builtin_amdgcn_wmma_f32_16x16x4_f32(
                             false, a, false, b, (short)0, acc[t], false, false);
            }
        }

        wait_async_lds();
        __syncthreads();
        buf ^= 1;
    }

    const float bv = bias[cc];
    #pragma unroll
    for (int t = 0; t < MT; ++t) {
        #pragma unroll
        for (int v = 0; v < 8; ++v) {
            int   r   = r0 + t * 16 + v + hi * 8;   // 16x16 f32 C/D layout
            float val = acc[t][v] + bv;
            if (MODE == 0) {
                val = val / (1.f + __expf(-val));   // SiLU
                if (cok) C[(size_t)r * NC + c] = val;
            } else {
                if (cok) {
                    C[(size_t)r * OUTC + c] = val;
                    if (c < LHALF) eimag[(size_t)r * LHALF + c]           = val;
                    else           ereal[(size_t)r * LHALF + (c - LHALF)] = val;
                }
            }
        }
    }
}

// ---------------------------------------------------------------------------
// Fused segmented softmax-attention pooling (online softmax, single pass):
//   h_G[g,c] = (sum_n h[n,c]*exp(att[n,c]-max)) / max(sum_n exp(att[n,c]-max), 1e-12)
// one block per graph, one thread per column; fully coalesced.
// ---------------------------------------------------------------------------
__global__ __launch_bounds__(256) void seg_pool_kernel(
    const float* __restrict__ att, const float* __restrict__ h,
    const int* __restrict__ seg, float* __restrict__ hG)
{
    const int g = blockIdx.x;
    const int c = threadIdx.x;              // 0..255
    const int s = seg[g], e = seg[g + 1];

    float m = -__builtin_inff(), ssum = 0.f, w = 0.f;
    for (int n = s; n < e; ++n) {
        float a  = att[(size_t)n * DIM + c];
        float hv = h  [(size_t)n * DIM + c];
        if (a > m) {                        // rescale running sums
            float sc = __expf(m - a);       // exp(-inf)=0 on first hit
            ssum *= sc; w *= sc; m = a;
        }
        float ev = __expf(a - m);
        ssum += ev;
        w    += hv * ev;
    }
    hG[(size_t)g * DIM + c] = w / fmaxf(ssum, 1e-12f);
}

// ---------------------------------------------------------------------------
extern "C" void kernel_launch(void* const* d_in, const int* in_sizes, int n_in,
                              void* d_out, int out_size, void* d_ws, size_t ws_size,
                              hipStream_t stream) {
    const float* h      = (const float*)d_in[0];
    const int*   ngi    = (const int*)  d_in[1];
    const float* W_pool = (const float*)d_in[2];
    const float* b_pool = (const float*)d_in[3];
    const float* W1     = (const float*)d_in[4];
    const float* b1     = (const float*)d_in[5];
    const float* W2     = (const float*)d_in[6];
    const float* b2     = (const float*)d_in[7];
    const float* W3     = (const float*)d_in[8];
    const float* b3     = (const float*)d_in[9];

    float* out   = (float*)d_out;
    float* eimag = out   + (size_t)NGRAPH * OUTC;
    float* ereal = eimag + (size_t)NGRAPH * LHALF;

    char* ws = (char*)d_ws;
    float* att = (float*)ws; ws += (size_t)N_NODES * DIM * sizeof(float);
    float* hG  = (float*)ws; ws += (size_t)NGRAPH  * DIM * sizeof(float);
    float* x1  = (float*)ws; ws += (size_t)NGRAPH  * HID * sizeof(float);
    float* x2  = (float*)ws; ws += (size_t)NGRAPH  * HID * sizeof(float);
    int*   seg = (int*)ws;

    // segment boundaries
    seg_bounds_kernel<<<(NGRAPH + 256) / 256, 256, 0, stream>>>(ngi, seg);

    // att = silu(h @ W_pool + b_pool)   [200000 x 256]
    gemm_wmma_kernel<DIM, DIM, 0, false>
        <<<dim3(N_NODES / 64, DIM / 128), 256, 0, stream>>>(
            h, W_pool, b_pool, att, nullptr, nullptr);

    // fused softmax-attention pooling -> h_G  [4096 x 256]
    seg_pool_kernel<<<NGRAPH, DIM, 0, stream>>>(att, h, seg, hG);

    // x1 = silu(h_G @ W1 + b1)  [4096 x 1024]
    gemm_wmma_kernel<DIM, HID, 0, false>
        <<<dim3(NGRAPH / 64, HID / 128), 256, 0, stream>>>(
            hG, W1, b1, x1, nullptr, nullptr);

    // x2 = silu(x1 @ W2 + b2)   [4096 x 1024]
    gemm_wmma_kernel<HID, HID, 0, false>
        <<<dim3(NGRAPH / 64, HID / 128), 256, 0, stream>>>(
            x1, W2, b2, x2, nullptr, nullptr);

    // out = x2 @ W3 + b3  [4096 x 4002], fused eps_imag/eps_real slice stores
    gemm_wmma_kernel<HID, OUTC, 1, true>
        <<<dim3(NGRAPH / 64, (OUTC + 127) / 128), 256, 0, stream>>>(
            x2, W3, b3, out, eimag, ereal);
}